// DecGreenNet_product_CP5_90967407329575
// MI455X (gfx1250) — compile-verified
//
#include <hip/hip_runtime.h>
#include <math.h>

typedef __attribute__((ext_vector_type(2))) float v2f;
typedef __attribute__((ext_vector_type(8))) float v8f;

#define NQ 128

// gfx1250 hardware V_TANH_F32 (TRANS32 op), with libm fallback.
__device__ __forceinline__ float dgn_tanh(float v) {
#if defined(__has_builtin)
#if __has_builtin(__builtin_amdgcn_tanhf)
  return __builtin_amdgcn_tanhf(v);
#else
  return tanhf(v);
#endif
#else
  return tanhf(v);
#endif
}

// ---------------------------------------------------------------------------
// Setup kernel (1 block, 1024 threads): collapses the 5 quadrature factors to
// two scalars each, builds rhs[1024], then w2r = Wx2 @ rhs (256) and
// b2r = bx2 . rhs (scalar).  Everything deterministic (fixed-order loops and
// tree reductions, no float atomics).
// ---------------------------------------------------------------------------
__global__ __launch_bounds__(1024)
void dgn_setup_kernel(const float* __restrict__ eq_param,
                      const float* __restrict__ qx0, const float* __restrict__ qx1,
                      const float* __restrict__ qx2, const float* __restrict__ qx3,
                      const float* __restrict__ qx4,
                      const float* __restrict__ Wx2, const float* __restrict__ bx2,
                      const float* __restrict__ Wq0, const float* __restrict__ bq0,
                      const float* __restrict__ Wq1, const float* __restrict__ bq1,
                      const float* __restrict__ Wq2, const float* __restrict__ bq2,
                      const float* __restrict__ Wq3, const float* __restrict__ bq3,
                      const float* __restrict__ Wq4, const float* __restrict__ bq4,
                      float* __restrict__ ws) {
  __shared__ float AB[5][2];     // per-axis (sum y*q, sum y)
  __shared__ float S[5][64];     // s_i[p*16+q]
  __shared__ float RHS[1024];
  __shared__ float RED[1024];
  const int t = (int)threadIdx.x;

  const float* qxs[5] = {qx0, qx1, qx2, qx3, qx4};
  const float* Wqs[5] = {Wq0, Wq1, Wq2, Wq3, Wq4};
  const float* bqs[5] = {bq0, bq1, bq2, bq3, bq4};

  if (t < 5) {
    const float eq = eq_param[0];
    const float* q = qxs[t];
    float a = 0.f, b = 0.f;
    for (int n = 0; n < NQ; ++n) {            // serial -> deterministic
      float xv = q[n];
      float y  = expf(-xv * xv * eq);         // r_func weight
      a += y * xv;
      b += y;
    }
    AB[t][0] = a;
    AB[t][1] = b;
  }
  __syncthreads();

  // s_i[k] = Wq_i[k] * A_i + bq_i[k] * B_i   (Linear(1->64) summed over quad pts)
  if (t < 320) {
    const int i = t >> 6, k = t & 63;
    S[i][k] = Wqs[i][k] * AB[i][0] + bqs[i][k] * AB[i][1];
  }
  __syncthreads();

  // rhs[b,d,f,m,v] = sum_x s0[b,x] s1[d,x] s2[f,x] s3[m,x] s4[v,x]
  {
    const int j = t;
    const int v = j & 3, m = (j >> 2) & 3, f = (j >> 4) & 3,
              d = (j >> 6) & 3, b = (j >> 8) & 3;
    float acc = 0.f;
    #pragma unroll
    for (int x = 0; x < 16; ++x)
      acc += S[0][b * 16 + x] * S[1][d * 16 + x] * S[2][f * 16 + x] *
             S[3][m * 16 + x] * S[4][v * 16 + x];
    RHS[j] = acc;
    RED[j] = bx2[j] * acc;                    // for b2r = bx2 . rhs
  }
  for (int s = 512; s > 0; s >>= 1) {         // deterministic tree reduce
    __syncthreads();
    if (t < s) RED[t] += RED[t + s];
  }
  __syncthreads();

  // w2r[h] = sum_j Wx2[h,j] * rhs[j]
  if (t < 256) {
    const float* row = Wx2 + (size_t)t * 1024;
    float acc = 0.f;
    for (int j = 0; j < 1024; ++j) acc += row[j] * RHS[j];
    ws[t] = acc;
  }
  if (t == 0) ws[256] = RED[0];
}

// ---------------------------------------------------------------------------
// Main kernel: one wave per 16 samples.
//   hidden+bias = X16x4_pad @ Wx1_4x256 + C(bias) via V_WMMA_F32_16X16X4_F32,
//   out[row] = sum_col tanh(hidden) * w2r[col] + b2r.
// Bias is fed through the WMMA C operand (D = A*B + C), so the epilogue per
// element is just v_tanh_f32 + v_fmac.
// Single-wave blocks -> EXEC is all-1s at every WMMA (ISA requirement).
// All fragment loads are branch-free (address math + selects, no EXEC edits).
// ---------------------------------------------------------------------------
__global__ __launch_bounds__(32)
void dgn_main_kernel(const float* __restrict__ x, const float* __restrict__ Wx1,
                     const float* __restrict__ bx1, const float* __restrict__ ws,
                     float* __restrict__ out, int N) {
  const int lane = (int)threadIdx.x & 31;
  const int half = lane >> 4;       // 0: holds K0,K1  1: holds K2,K3(pad)
  const int lmod = lane & 15;
  const long rowBase = (long)blockIdx.x * 16;
  const float kmask = (half == 0) ? 1.0f : 0.0f;   // zeroes the K=3 pad slot

  // B fragments: Wx1 is [3][256] row-major; tile tl covers cols 16*tl..16*tl+15.
  // Lane layout (ISA 7.12.2): lanes 0-15 comp0/1 = K0/K1, lanes 16-31 = K2/K3.
  v2f B[16];
  float bb[16], wv[16];
  #pragma unroll
  for (int tl = 0; tl < 16; ++tl) {
    const int col = tl * 16 + lmod;
    B[tl][0] = Wx1[half * 512 + col];
    B[tl][1] = Wx1[256 + col] * kmask;
    bb[tl]   = bx1[col];              // bias, same for every row of the tile
    wv[tl]   = ws[col];               // w2r
  }
  const float b2r = ws[256];

  // A fragment: row = rowBase + lmod for BOTH halves (16x4, K split by half).
  long r = rowBase + lmod;
  if (r > (long)N - 1) r = (long)N - 1;   // clamp tail (dup rows, never stored)
  const float* xr = x + r * 3;
  v2f A;
  A[0] = xr[2 * half];                    // K0 or K2 (always in bounds)
  A[1] = xr[1] * kmask;                   // K1, or 0 pad for K=3

  float acc[8];
  #pragma unroll
  for (int v = 0; v < 8; ++v) acc[v] = 0.f;

  #pragma unroll
  for (int tl = 0; tl < 16; ++tl) {
    // C layout matches D: comp v, lanes0-15 -> row v, lanes16-31 -> row v+8,
    // N = lane%16.  Bias depends only on col => same value in all 8 comps.
    v8f c;
    #pragma unroll
    for (int v = 0; v < 8; ++v) c[v] = bb[tl];
    // D = A(16x4) x B(4x16) + bias : fp32 WMMA
    c = __builtin_amdgcn_wmma_f32_16x16x4_f32(false, A, false, B[tl],
                                              (short)0, c, false, false);
    #pragma unroll
    for (int v = 0; v < 8; ++v)
      acc[v] += dgn_tanh(c[v]) * wv[tl];
  }

  // Reduce over the 16 columns held by each 16-lane half.
  #pragma unroll
  for (int v = 0; v < 8; ++v) {
    float a = acc[v];
    a += __shfl_xor(a, 1, 16);
    a += __shfl_xor(a, 2, 16);
    a += __shfl_xor(a, 4, 16);
    a += __shfl_xor(a, 8, 16);
    acc[v] = a;
  }

  if (lmod == 0) {                   // lane0 -> rows +0..7, lane16 -> rows +8..15
    #pragma unroll
    for (int v = 0; v < 8; ++v) {
      const long row = rowBase + (long)half * 8 + v;
      if (row < N) out[row] = acc[v] + b2r;
    }
  }
}

// ---------------------------------------------------------------------------
extern "C" void kernel_launch(void* const* d_in, const int* in_sizes, int n_in,
                              void* d_out, int out_size, void* d_ws, size_t ws_size,
                              hipStream_t stream) {
  const float* input = (const float*)d_in[0];   // [N,3]
  const float* eq    = (const float*)d_in[1];   // [1]
  const float* qx0   = (const float*)d_in[2];
  const float* qx1   = (const float*)d_in[3];
  const float* qx2   = (const float*)d_in[4];
  const float* qx3   = (const float*)d_in[5];
  const float* qx4   = (const float*)d_in[6];
  const float* Wx1   = (const float*)d_in[7];   // [3,256]
  const float* bx1   = (const float*)d_in[8];   // [256]
  const float* Wx2   = (const float*)d_in[9];   // [256,1024]
  const float* bx2   = (const float*)d_in[10];  // [1024]
  const float* Wq0   = (const float*)d_in[11];
  const float* bq0   = (const float*)d_in[12];
  const float* Wq1   = (const float*)d_in[13];
  const float* bq1   = (const float*)d_in[14];
  const float* Wq2   = (const float*)d_in[15];
  const float* bq2   = (const float*)d_in[16];
  const float* Wq3   = (const float*)d_in[17];
  const float* bq3   = (const float*)d_in[18];
  const float* Wq4   = (const float*)d_in[19];
  const float* bq4   = (const float*)d_in[20];

  float* ws  = (float*)d_ws;    // ws[0..255] = w2r, ws[256] = b2r
  float* out = (float*)d_out;
  const int N = in_sizes[0] / 3;

  dgn_setup_kernel<<<1, 1024, 0, stream>>>(eq, qx0, qx1, qx2, qx3, qx4,
                                           Wx2, bx2,
                                           Wq0, bq0, Wq1, bq1, Wq2, bq2,
                                           Wq3, bq3, Wq4, bq4, ws);

  const int nWaves = (N + 15) / 16;             // one wave per 16 rows
  dgn_main_kernel<<<nWaves, 32, 0, stream>>>(input, Wx1, bx1, ws, out, N);
}